// SelectExperts_60447369724371
// MI455X (gfx1250) — compile-verified
//
#include <hip/hip_runtime.h>
#include <hip/hip_bf16.h>
#include <math.h>

typedef __attribute__((ext_vector_type(2))) float v2f;
typedef __attribute__((ext_vector_type(8))) float v8f;

#define BH        128      // B*H rows
#define MKEYS     65536
#define DDIM      256
#define NTOP      16
#define NRAND     4

// ---------------------------------------------------------------------------
// Kernel 1: logits = Q (128x256) * K^T (256x65536), fp32 via WMMA 16x16x4.
// One wave handles one 16-key tile across all 8 query tiles; B-frag is loaded
// once per K-step and reused by 8 WMMAs.
// A-frag layout (ISA 7.12.2): lane L -> m=L%16, VGPR pair = Q[m][k0+2*(L>>4)..+1]
// B-frag layout:              lane L -> n=L%16, VGPR pair = K[n][k0+2*(L>>4)..+1]
// C/D layout: acc[r] on lane L -> C[m = r + 8*(L>>4)][n = L%16]
// ---------------------------------------------------------------------------
__global__ __launch_bounds__(256) void gemm_logits_kernel(
    const float* __restrict__ q, const float* __restrict__ kmat,
    float* __restrict__ logits)
{
  const int lane    = threadIdx.x & 31;
  const int wave    = threadIdx.x >> 5;
  const int keyTile = blockIdx.x * 8 + wave;            // 0..4095
  const int n       = lane & 15;
  const int kh      = lane >> 4;                        // 0 or 1
  const float* kptr = kmat + (size_t)(keyTile * 16 + n) * DDIM;

  v8f acc[8] = {};

  for (int k0 = 0; k0 < DDIM; k0 += 4) {
    v2f b = *(const v2f*)(kptr + k0 + 2 * kh);
#pragma unroll
    for (int t = 0; t < 8; ++t) {
      const float* qptr = q + (size_t)(t * 16 + n) * DDIM;
      v2f a = *(const v2f*)(qptr + k0 + 2 * kh);
      acc[t] = __builtin_amdgcn_wmma_f32_16x16x4_f32(
          /*neg_a=*/false, a, /*neg_b=*/false, b,
          /*c_mod=*/(short)0, acc[t], /*reuse_a=*/false, /*reuse_b=*/false);
    }
  }

#pragma unroll
  for (int t = 0; t < 8; ++t) {
#pragma unroll
    for (int r = 0; r < 8; ++r) {
      int m = t * 16 + r + 8 * kh;
      logits[(size_t)m * MKEYS + keyTile * 16 + n] = acc[t][r];
    }
  }
}

// ---------------------------------------------------------------------------
// Kernel 2: per-row top-16 by iterative argmax extraction (L2-resident scans).
// Ties broken toward lower index (matches jax.lax.top_k); extracted entry is
// poisoned to -inf so rand gather positions (never top) stay untouched.
// ---------------------------------------------------------------------------
__global__ __launch_bounds__(1024) void topk_kernel(
    float* __restrict__ logits, int* __restrict__ top_idx,
    float* __restrict__ top_val)
{
  __shared__ float sval[1024];
  __shared__ int   sidx[1024];
  const int row = blockIdx.x;
  float* rowp = logits + (size_t)row * MKEYS;
  const int tid = threadIdx.x;

  for (int it = 0; it < NTOP; ++it) {
    float best = -INFINITY;
    int   bidx = MKEYS;
    for (int i = tid; i < MKEYS; i += 1024) {
      float v = rowp[i];
      if (v > best || (v == best && i < bidx)) { best = v; bidx = i; }
    }
    sval[tid] = best; sidx[tid] = bidx;
    __syncthreads();
    for (int s = 512; s > 0; s >>= 1) {
      if (tid < s) {
        float ov = sval[tid + s]; int oi = sidx[tid + s];
        if (ov > sval[tid] || (ov == sval[tid] && oi < sidx[tid])) {
          sval[tid] = ov; sidx[tid] = oi;
        }
      }
      __syncthreads();
    }
    if (tid == 0) {
      top_val[row * NTOP + it] = sval[0];
      top_idx[row * NTOP + it] = sidx[0];
      rowp[sidx[0]] = -INFINITY;            // exclude from next extraction
    }
    __syncthreads();
  }
}

// ---------------------------------------------------------------------------
// Kernel 3: global exclusion mask over M from all 128*16 top indices.
// ---------------------------------------------------------------------------
__global__ __launch_bounds__(1024) void mask_kernel(
    const int* __restrict__ top_idx, int* __restrict__ mask)
{
  for (int i = threadIdx.x; i < MKEYS; i += 1024) mask[i] = 0;
  __syncthreads();
  for (int i = threadIdx.x; i < BH * NTOP; i += 1024) mask[top_idx[i]] = 1;
}

// Deterministic hash -> uniform [0,1) (stand-in for the JAX threefry stream).
__device__ __forceinline__ float hash_uniform(unsigned int i) {
  unsigned int h = i * 2654435761u;
  h ^= h >> 16; h *= 0x85ebca6bu;
  h ^= h >> 13; h *= 0xc2b2ae35u;
  h ^= h >> 16;
  return (float)h * (1.0f / 4294967296.0f);
}

// ---------------------------------------------------------------------------
// Kernel 4: pick NRAND smallest hash-uniforms among non-excluded indices.
// ---------------------------------------------------------------------------
__global__ __launch_bounds__(1024) void rand_kernel(
    const int* __restrict__ mask, int* __restrict__ rand_idx)
{
  __shared__ float sval[1024];
  __shared__ int   sidx[1024];
  __shared__ int   chosen[NRAND];
  const int tid = threadIdx.x;

  for (int it = 0; it < NRAND; ++it) {
    float best = INFINITY;
    int   bidx = MKEYS;
    for (int i = tid; i < MKEYS; i += 1024) {
      if (mask[i]) continue;
      bool skip = false;
      for (int j = 0; j < it; ++j) if (chosen[j] == i) skip = true;
      if (skip) continue;
      float u = hash_uniform((unsigned int)i);
      if (u < best || (u == best && i < bidx)) { best = u; bidx = i; }
    }
    sval[tid] = best; sidx[tid] = bidx;
    __syncthreads();
    for (int s = 512; s > 0; s >>= 1) {
      if (tid < s) {
        float ov = sval[tid + s]; int oi = sidx[tid + s];
        if (ov < sval[tid] || (ov == sval[tid] && oi < sidx[tid])) {
          sval[tid] = ov; sidx[tid] = oi;
        }
      }
      __syncthreads();
    }
    if (tid == 0) { chosen[it] = sidx[0]; rand_idx[it] = sidx[0]; }
    __syncthreads();
  }
}

// ---------------------------------------------------------------------------
// Kernel 5: gather rand logits, 20-way softmax, emit (indices-as-float, probs).
// ---------------------------------------------------------------------------
__global__ __launch_bounds__(128) void finalize_kernel(
    const float* __restrict__ logits, const int* __restrict__ top_idx,
    const float* __restrict__ top_val, const int* __restrict__ rand_idx,
    float* __restrict__ out)
{
  const int row = threadIdx.x;
  if (row >= BH) return;
  float lg[NTOP + NRAND];
  int   id[NTOP + NRAND];
#pragma unroll
  for (int t = 0; t < NTOP; ++t) {
    lg[t] = top_val[row * NTOP + t];
    id[t] = top_idx[row * NTOP + t];
  }
#pragma unroll
  for (int j = 0; j < NRAND; ++j) {
    int ri = rand_idx[j];
    id[NTOP + j] = ri;
    lg[NTOP + j] = logits[(size_t)row * MKEYS + ri];
  }
  float mx = lg[0];
#pragma unroll
  for (int t = 1; t < NTOP + NRAND; ++t) mx = fmaxf(mx, lg[t]);
  float e[NTOP + NRAND];
  float sum = 0.0f;
#pragma unroll
  for (int t = 0; t < NTOP + NRAND; ++t) { e[t] = __expf(lg[t] - mx); sum += e[t]; }
  const float inv = 1.0f / sum;
#pragma unroll
  for (int t = 0; t < NTOP + NRAND; ++t) {
    out[row * (NTOP + NRAND) + t]                      = (float)id[t];
    out[BH * (NTOP + NRAND) + row * (NTOP + NRAND) + t] = e[t] * inv;
  }
}

extern "C" void kernel_launch(void* const* d_in, const int* in_sizes, int n_in,
                              void* d_out, int out_size, void* d_ws, size_t ws_size,
                              hipStream_t stream) {
  const float* q    = (const float*)d_in[0];   // (8,16,256)
  const float* kmat = (const float*)d_in[1];   // (65536,256)
  float* out        = (float*)d_out;           // 2560 idx + 2560 probs

  // Workspace layout
  char* ws = (char*)d_ws;
  float* logits  = (float*)ws;                                   // 128*65536*4 = 33.5 MB
  size_t off     = (size_t)BH * MKEYS * sizeof(float);
  float* top_val = (float*)(ws + off); off += BH * NTOP * sizeof(float);
  int*   top_idx = (int*)  (ws + off); off += BH * NTOP * sizeof(int);
  int*   mask    = (int*)  (ws + off); off += MKEYS * sizeof(int);
  int*   rand_idx= (int*)  (ws + off);

  // 1) fp32 WMMA GEMM: 4096 key tiles, 8 waves/block
  gemm_logits_kernel<<<MKEYS / 16 / 8, 256, 0, stream>>>(q, kmat, logits);
  // 2) top-16 per row
  topk_kernel<<<BH, 1024, 0, stream>>>(logits, top_idx, top_val);
  // 3) exclusion mask
  mask_kernel<<<1, 1024, 0, stream>>>(top_idx, mask);
  // 4) 4 random non-excluded indices
  rand_kernel<<<1, 1024, 0, stream>>>(mask, rand_idx);
  // 5) gather + softmax + emit
  finalize_kernel<<<1, 128, 0, stream>>>(logits, top_idx, top_val, rand_idx, out);
}